// YOLOLoss2_77910706749528
// MI455X (gfx1250) — compile-verified
//
#include <hip/hip_runtime.h>
#include <hip/hip_bf16.h>
#include <cstdint>
#include <cstddef>

// YOLO loss, MI455X (gfx1250). Only masked cells contribute to the loss, so we
// gather <=256 target cells per scale instead of sweeping 155MB of activations.
// CDNA5 paths: async global->LDS staging (ASYNCcnt) + v_wmma_f32_16x16x4_f32
// block reductions (exact f32 sums, deterministic).

#define NCLS 80
#define REGM 16
#define CCH  144      // 4*REG_MAX + NC
#define NTMAX 256

typedef __attribute__((ext_vector_type(2))) float v2f;
typedef __attribute__((ext_vector_type(8))) float v8f;

#define AS1 __attribute__((address_space(1)))
#define AS3 __attribute__((address_space(3)))

#if defined(__HIP_DEVICE_COMPILE__) && __has_builtin(__builtin_amdgcn_global_load_async_to_lds_b32)
#define HAVE_ASYNC_G2L 1
#endif

__device__ __forceinline__ void g2l_b32(const float* g, float* l) {
#if defined(HAVE_ASYNC_G2L)
    __builtin_amdgcn_global_load_async_to_lds_b32(
        (AS1 int*)g, (AS3 int*)l, 0, 0);
#else
    *l = *g;
#endif
}

__device__ __forceinline__ void wait_g2l() {
#if defined(HAVE_ASYNC_G2L)
#if __has_builtin(__builtin_amdgcn_s_wait_asynccnt)
    __builtin_amdgcn_s_wait_asynccnt(0);
#else
    asm volatile("s_wait_asynccnt 0" ::: "memory");
#endif
#endif
}

// Exact f32 wave(32)-sum via V_WMMA_F32_16X16X4_F32 with B = ones.
// A lane layout (ISA 7.12.2): lanes 0-15 supply rows M=0..15 at K=0,1 (v0,v1),
// lanes 16-31 supply K=2,3.  With A={q,0}, B=ones:  D[m][n] = q_m + q_{m+16}.
// Per-lane sum of the 8 D VGPRs gives the half-sum for that lane's 16-group;
// shfl_xor(16) completes the full 32-lane sum on every lane.
__device__ __forceinline__ float wave_reduce_wmma(float q) {
    v2f a; a.x = q;   a.y = 0.0f;
    v2f b; b.x = 1.0f; b.y = 1.0f;
    v8f c = {0.f, 0.f, 0.f, 0.f, 0.f, 0.f, 0.f, 0.f};
    v8f d = __builtin_amdgcn_wmma_f32_16x16x4_f32(
        /*neg_a=*/false, a, /*neg_b=*/false, b,
        /*c_mod=*/(short)0, c, /*reuse_a=*/false, /*reuse_b=*/false);
    float s = d[0] + d[1] + d[2] + d[3] + d[4] + d[5] + d[6] + d[7];
    s += __shfl_xor(s, 16, 32);
    return s;
}

__global__ __launch_bounds__(256) void yolo_loss_scale(
    const float* __restrict__ p0, const float* __restrict__ p1,
    const float* __restrict__ p2, const float* __restrict__ tgt,
    float* __restrict__ acc, int nt)
{
    __shared__ float s_tgt[NTMAX * 6];
    __shared__ int   s_cell[NTMAX];
    __shared__ int   s_cls[NTMAX];
    __shared__ float s_red[8][4];

    const int tid   = threadIdx.x;
    const int scale = blockIdx.x;
    const float* pred = (scale == 0) ? p0 : (scale == 1) ? p1 : p2;
    const int H  = (scale == 0) ? 80 : (scale == 1) ? 40 : 20;
    const int W  = H;
    const int HW = H * W;

    // Stage target table into LDS via async global->LDS (ASYNCcnt path).
    if (tid < nt) {
#pragma unroll
        for (int k = 0; k < 6; ++k)
            g2l_b32(tgt + tid * 6 + k, &s_tgt[tid * 6 + k]);
    }
    wait_g2l();
    __syncthreads();

    float x = 0.f, y = 0.f, wb = 0.f, hb = 0.f;
    int img = 0;
    if (tid < nt) {
        img    = (int)s_tgt[tid * 6 + 0];
        int cl = (int)s_tgt[tid * 6 + 1];
        x  = s_tgt[tid * 6 + 2];  y  = s_tgt[tid * 6 + 3];
        wb = s_tgt[tid * 6 + 4];  hb = s_tgt[tid * 6 + 5];
        int gx = min((int)(x * (float)W), W - 1);
        int gy = min((int)(y * (float)H), H - 1);
        s_cell[tid] = (img * H + gy) * W + gx;
        s_cls[tid]  = cl;
    } else if (tid < NTMAX) {
        s_cell[tid] = -1; s_cls[tid] = 0;
    }
    __syncthreads();

    float boxP = 0.f, clsP = 0.f, dflP = 0.f, cntP = 0.f;

    if (tid < nt) {
        const int myCell = s_cell[tid];
        bool win = true;                 // last duplicate index wins (JAX .set)
        unsigned cm0 = 0, cm1 = 0, cm2 = 0;  // union of class one-hots per cell
        for (int j = 0; j < nt; ++j) {
            if (s_cell[j] == myCell) {
                if (j > tid) win = false;
                int c = s_cls[j];
                unsigned bit = 1u << (c & 31);
                if (c < 32) cm0 |= bit; else if (c < 64) cm1 |= bit; else cm2 |= bit;
            }
        }
        if (win) {
            int gx = myCell % W;
            int gy = (myCell / W) % H;
            const float* base = pred + (size_t)img * (size_t)CCH * (size_t)HW
                                     + (size_t)gy * W + gx;
            __builtin_prefetch(base, 0, 0);

            // ---- DFL + distribution expectation (softmax over 16 bins x 4) ----
            float lt[4] = { x - wb * 0.5f, y - hb * 0.5f, x + wb * 0.5f, y + hb * 0.5f };
            float px[4];
            float dfl = 0.f;
#pragma unroll
            for (int sd = 0; sd < 4; ++sd) {
                float z[REGM];
#pragma unroll
                for (int j = 0; j < REGM; ++j)
                    z[j] = base[(size_t)(sd * REGM + j) * HW];
                float m = z[0];
#pragma unroll
                for (int j = 1; j < REGM; ++j) m = fmaxf(m, z[j]);
                float ssum = 0.f, dot = 0.f;
#pragma unroll
                for (int j = 0; j < REGM; ++j) {
                    float e = expf(z[j] - m);
                    ssum += e; dot += e * (float)j;
                }
                px[sd] = dot / ssum;
                float t = lt[sd] * (float)(REGM - 1);
                t = fminf(fmaxf(t, 0.f), (float)(REGM - 1) - 1e-6f);
                int   tl = (int)t;
                int   tr = min(tl + 1, REGM - 1);
                float wr = t - (float)tl;
                float wl = 1.f - wr;
                // re-fetch dynamic bins from global (L0-hot) to avoid scratch
                float ztl = base[(size_t)(sd * REGM + tl) * HW];
                float ztr = base[(size_t)(sd * REGM + tr) * HW];
                float pl = fminf(fmaxf(expf(ztl - m) / ssum, 1e-6f), 1.f - 1e-6f);
                float pr = fminf(fmaxf(expf(ztr - m) / ssum, 1e-6f), 1.f - 1e-6f);
                dfl += -logf(pl) * wl - logf(pr) * wr;
            }

            // ---- CIoU (reference feeds (x,y,w,h) directly as corner coords) ----
            const float eps = 1e-7f;
            float b1x1 = px[0], b1y1 = px[1], b1x2 = px[2], b1y2 = px[3];
            float b2x1 = x, b2y1 = y, b2x2 = wb, b2y2 = hb;
            float iw = fmaxf(fminf(b1x2, b2x2) - fmaxf(b1x1, b2x1), 0.f);
            float ih = fmaxf(fminf(b1y2, b2y2) - fmaxf(b1y1, b2y1), 0.f);
            float inter = iw * ih;
            float w1 = b1x2 - b1x1, h1 = b1y2 - b1y1;
            float w2 = b2x2 - b2x1, h2 = b2y2 - b2y1;
            float uni = w1 * h1 + w2 * h2 - inter + eps;
            float iou = inter / uni;
            float cw = fmaxf(b1x2, b2x2) - fminf(b1x1, b2x1);
            float chh = fmaxf(b1y2, b2y2) - fminf(b1y1, b2y1);
            float c2 = cw * cw + chh * chh + eps;
            float dx = b2x1 + b2x2 - b1x1 - b1x2;
            float dy = b2y1 + b2y2 - b1y1 - b1y2;
            float rho2 = (dx * dx + dy * dy) * 0.25f;
            const float k4pi2 = 4.f / (3.14159265358979323846f * 3.14159265358979323846f);
            float dv = atanf(w2 / h2) - atanf(w1 / h1);
            float v  = k4pi2 * dv * dv;
            float alpha = v / (v - iou + 1.f + eps);
            float ciou  = iou - (rho2 / c2 + v * alpha);

            // ---- class BCE over 80 logits ----
            float bce = 0.f;
            for (int c = 0; c < NCLS; ++c) {
                float zc = base[(size_t)(4 * REGM + c) * HW];
                unsigned mword = (c < 32) ? cm0 : (c < 64) ? cm1 : cm2;
                float tc = (float)((mword >> (c & 31)) & 1u);
                bce += fmaxf(zc, 0.f) - zc * tc + log1pf(expf(-fabsf(zc)));
            }

            boxP = 1.f - ciou;
            clsP = bce;
            dflP = dfl;
            cntP = 1.f;
        }
    }

    // ---- block reduction: wave-level WMMA sums, then 8-wave combine ----
    float tb = wave_reduce_wmma(boxP);
    float tc = wave_reduce_wmma(clsP);
    float td = wave_reduce_wmma(dflP);
    float tn = wave_reduce_wmma(cntP);
    int wv = tid >> 5;
    if ((tid & 31) == 0) {
        s_red[wv][0] = tb; s_red[wv][1] = tc; s_red[wv][2] = td; s_red[wv][3] = tn;
    }
    __syncthreads();
    if (tid == 0) {
        float b = 0.f, c = 0.f, d = 0.f, n = 0.f;
#pragma unroll
        for (int w8 = 0; w8 < 8; ++w8) {
            b += s_red[w8][0]; c += s_red[w8][1]; d += s_red[w8][2]; n += s_red[w8][3];
        }
        acc[scale * 4 + 0] = b;
        acc[scale * 4 + 1] = c;
        acc[scale * 4 + 2] = d;
        acc[scale * 4 + 3] = n;
    }
}

__global__ void yolo_loss_finalize(const float* __restrict__ acc,
                                   float* __restrict__ out)
{
    if (threadIdx.x == 0 && blockIdx.x == 0) {
        float box = 0.f, cl = 0.f, dfl = 0.f;
#pragma unroll
        for (int s = 0; s < 3; ++s) {
            float cnt = fmaxf(acc[s * 4 + 3], 1.f);
            box += acc[s * 4 + 0] / cnt;
            cl  += acc[s * 4 + 1] / (cnt * (float)NCLS);
            dfl += acc[s * 4 + 2] / (cnt * 4.f);
        }
        box *= 10.0f; cl *= 0.5f; dfl *= 2.0f;
        out[0] = box; out[1] = cl; out[2] = dfl; out[3] = box + cl + dfl;
    }
}

extern "C" void kernel_launch(void* const* d_in, const int* in_sizes, int n_in,
                              void* d_out, int out_size, void* d_ws, size_t ws_size,
                              hipStream_t stream) {
    const float* p0 = (const float*)d_in[0];
    const float* p1 = (const float*)d_in[1];
    const float* p2 = (const float*)d_in[2];
    const float* tg = (const float*)d_in[3];
    int nt = in_sizes[3] / 6;
    if (nt > NTMAX) nt = NTMAX;
    float* acc = (float*)d_ws;   // 12 floats, fully overwritten each call
    yolo_loss_scale<<<dim3(3), dim3(256), 0, stream>>>(p0, p1, p2, tg, acc, nt);
    yolo_loss_finalize<<<dim3(1), dim3(32), 0, stream>>>(acc, (float*)d_out);
}